// BlockQuantizer_65369402245365
// MI455X (gfx1250) — compile-verified
//
#include <hip/hip_runtime.h>
#include <hip/hip_bf16.h>

#define ROW_LEN   16384               // elements per block-quantization row
#define THREADS   256                 // 8 wave32 per workgroup
#define VPT       (ROW_LEN / 4 / THREADS)   // 16 float4 per thread

typedef float         v4f  __attribute__((ext_vector_type(4)));
typedef unsigned int  u32x4 __attribute__((ext_vector_type(4)));
typedef int           i32x4 __attribute__((ext_vector_type(4)));
typedef int           i32x8 __attribute__((ext_vector_type(8)));

#if defined(__has_builtin)
#  if __has_builtin(__builtin_amdgcn_tensor_load_to_lds) && __has_builtin(__builtin_amdgcn_s_wait_tensorcnt)
#    define USE_TDM 1
#  else
#    define USE_TDM 0
#  endif
#else
#  define USE_TDM 0
#endif

__global__ __launch_bounds__(THREADS, 2)
void bq_rowquant_kernel(const float* __restrict__ x,
                        const float* __restrict__ noise,
                        float* __restrict__ out) {
  // Row staged once into LDS via TDM; both passes (max + quantize) read LDS.
  __shared__ v4f   sxv[ROW_LEN / 4];       // 64 KB
  __shared__ float sred[10];               // 8 wave partials + scale + inv_scale

  const int tid = threadIdx.x;
  const long long row = blockIdx.x;
  const v4f* __restrict__ xrow = reinterpret_cast<const v4f*>(x)     + row * (ROW_LEN / 4);
  const v4f* __restrict__ nrow = reinterpret_cast<const v4f*>(noise) + row * (ROW_LEN / 4);
  v4f*       __restrict__ orow = reinterpret_cast<v4f*>(out)         + row * (ROW_LEN / 4);

  float m = 0.0f;

#if USE_TDM
  // ---- Phase 0: TDM DMA of the whole row (64 KB) global -> LDS ----
  if (tid < 32) {  // one wave issues the tensor op (EXEC ignored by TDM)
    unsigned long long ga = (unsigned long long)(const void*)xrow;
    unsigned lds_off = (unsigned)(unsigned long long)(void*)&sxv[0]; // flat LDS addr low 32b == LDS offset

    // D# group 0: count=1 (valid, user mode), lds_addr, global_addr[56:0], type=2 ("image")
    u32x4 g0;
    g0[0] = 1u;
    g0[1] = lds_off;
    g0[2] = (unsigned)(ga & 0xFFFFFFFFull);
    g0[3] = (unsigned)((ga >> 32) & 0x01FFFFFFull) | (2u << 30);

    // D# group 1: wg_mask=0 (not in cluster), data_size=2 (4 bytes),
    // tensor_dim0 = tile_dim0 = 16384, tensor_dim1 = tile_dim1 = 1,
    // tensor_dim0_stride = 16384, no padding / iterate / atomic-barrier.
    i32x8 g1;
    g1[0] = (int)(2u << 16);                          // data_size = 4B
    g1[1] = (int)((unsigned)(ROW_LEN & 0xFFFF) << 16);// tensor_dim0[15:0] @ bits 63:48
    g1[2] = (int)(1u << 16);                          // tensor_dim0[31:16]=0, tensor_dim1[15:0]=1
    g1[3] = (int)((unsigned)(ROW_LEN & 0xFFFF) << 16);// tensor_dim1[31:16]=0, tile_dim0=16384
    g1[4] = 1;                                        // tile_dim1=1, tile_dim2=0
    g1[5] = ROW_LEN;                                  // tensor_dim0_stride[31:0]
    g1[6] = 0;                                        // stride hi / dim1_stride lo
    g1[7] = 0;                                        // dim1_stride hi

    i32x4 gz4 = {0, 0, 0, 0};                         // groups 2/3: 2D tensor, all zero
    i32x8 gz8 = {0, 0, 0, 0, 0, 0, 0, 0};             // clang-23 6-arg form: extra zero group
    __builtin_amdgcn_tensor_load_to_lds(g0, g1, gz4, gz4, gz8, 0);
    __builtin_amdgcn_s_wait_tensorcnt(0);
  }
  __syncthreads();

  // ---- Phase 1: row max(|x|) from LDS ----
  for (int i = 0; i < VPT; ++i) {
    v4f v = sxv[tid + i * THREADS];
    m = fmaxf(m, fmaxf(fmaxf(__builtin_fabsf(v.x), __builtin_fabsf(v.y)),
                       fmaxf(__builtin_fabsf(v.z), __builtin_fabsf(v.w))));
  }
#else
  // ---- Fallback: stage via global_load_b128 + ds_store_b128, max from registers ----
  for (int i = 0; i < VPT; ++i) {
    v4f v = xrow[tid + i * THREADS];
    sxv[tid + i * THREADS] = v;
    m = fmaxf(m, fmaxf(fmaxf(__builtin_fabsf(v.x), __builtin_fabsf(v.y)),
                       fmaxf(__builtin_fabsf(v.z), __builtin_fabsf(v.w))));
  }
#endif

  // ---- Wave32 shuffle reduction, then cross-wave via LDS ----
  for (int off = 16; off > 0; off >>= 1)
    m = fmaxf(m, __shfl_xor(m, off, 32));
  if ((tid & 31) == 0) sred[tid >> 5] = m;
  __syncthreads();

  if (tid == 0) {
    float bm = sred[0];
    for (int w = 1; w < THREADS / 32; ++w) bm = fmaxf(bm, sred[w]);
    // floor(log2(bm)) computed exactly: frexp gives bm = f*2^e, f in [0.5,1) -> floor(log2)=e-1
    int e = 0;
    (void)frexpf(bm, &e);
    int me = e - 1;
    if (bm == 0.0f) me = -128;            // log2(0) = -inf -> clipped to -2^7
    if (me < -128) me = -128;
    if (me >  127) me =  127;
    sred[8] = ldexpf(1.0f, 6 - me);       // scale = 2^(-max_exp + bits-2), exact (inf if overflow, matches ref)
    sred[9] = ldexpf(1.0f, me - 6);       // 1/scale, exact power of two
  }
  __syncthreads();

  const float scale = sred[8];
  const float inv   = sred[9];

  // ---- Phase 2: quantize from LDS, noise streamed from global, b128 stores ----
  for (int i = 0; i < VPT; ++i) {
    const int idx = tid + i * THREADS;
    v4f v = sxv[idx];
    v4f n = nrow[idx];
    v4f r;
    // x*scale is exact (scale = 2^k), so fmaf == mul+add here; rintf = round-half-even (jnp.round)
    r.x = rintf(fminf(fmaxf(fmaf(v.x, scale, n.x - 0.5f), -128.0f), 127.0f)) * inv;
    r.y = rintf(fminf(fmaxf(fmaf(v.y, scale, n.y - 0.5f), -128.0f), 127.0f)) * inv;
    r.z = rintf(fminf(fmaxf(fmaf(v.z, scale, n.z - 0.5f), -128.0f), 127.0f)) * inv;
    r.w = rintf(fminf(fmaxf(fmaf(v.w, scale, n.w - 0.5f), -128.0f), 127.0f)) * inv;
    orow[idx] = r;
  }
}

extern "C" void kernel_launch(void* const* d_in, const int* in_sizes, int n_in,
                              void* d_out, int out_size, void* d_ws, size_t ws_size,
                              hipStream_t stream) {
  const float* x     = (const float*)d_in[0];
  const float* noise = (const float*)d_in[1];
  float* out         = (float*)d_out;
  (void)n_in; (void)d_ws; (void)ws_size; (void)out_size;

  const int rows = in_sizes[0] / ROW_LEN;   // 4096 for the reference shapes
  bq_rowquant_kernel<<<dim3(rows), dim3(THREADS), 0, stream>>>(x, noise, out);
}